// TemporalFlowCell_73478300500531
// MI455X (gfx1250) — compile-verified
//
#include <hip/hip_runtime.h>
#include <hip/hip_bf16.h>
#include <math.h>

// Problem dims (fixed by the reference)
#define Tt 2048
#define Bb 16
#define Dd 1024
#define Kk 256

// LDS padding (in floats)
#define RPAD 264   // R row stride: 264%64==8 (lanes conflict-free), 2*264%64==16 (halves disjoint)
#define NPAD 260   // nr/ni row stride: even (b64-aligned), 260%64==4 -> conflict-free A frags
#define APAD 34    // GEMM A-tile row stride (16 x 32 tile), even for float2 loads
#define BPAD 257   // GEMM B-tile row stride (32 x 256 tile)

typedef __attribute__((ext_vector_type(2))) float v2f;
typedef __attribute__((ext_vector_type(8))) float v8f;
typedef __attribute__((ext_vector_type(4))) unsigned int u32x4;
typedef __attribute__((ext_vector_type(8))) int i32x8;
typedef __attribute__((ext_vector_type(4))) int i32x4;

// D = A(16x4) * B(4x16) + C, fp32, wave32.
// 8-arg pattern: (neg_a, A, neg_b, B, c_mod, C, reuse_a, reuse_b)
__device__ __forceinline__ v8f wmma4(v2f a, v2f b, v8f c) {
  return __builtin_amdgcn_wmma_f32_16x16x4_f32(false, a, false, b, (short)0, c,
                                               false, false);
}

// ---------------------------------------------------------------------------
// TDM: DMA a contiguous row of 4096 f32 (16 KB) from global into LDS.
// D# group0: count=1, lds_addr, global_addr[56:0], type=2 ("image").
// D# group1: data_size=4B; tensor_dim0=tile_dim0=4096; tensor_dim1=tile_dim1=1;
//            tensor_dim0_stride=4096; no pad/iterate/multicast.
// ---------------------------------------------------------------------------
__device__ __forceinline__ void tdm_load_row4096(const float* gsrc, float* ldst) {
  unsigned lds_off = (unsigned)(reinterpret_cast<uintptr_t>(ldst));  // flat LDS low32 = LDS offset
  unsigned long long ga = (unsigned long long)(reinterpret_cast<uintptr_t>(gsrc));
  u32x4 g0;
  g0[0] = 1u;                                                 // count=1, user D#
  g0[1] = lds_off;                                            // lds_addr (bytes)
  g0[2] = (unsigned)ga;                                       // global_addr[31:0]
  g0[3] = (unsigned)((ga >> 32) & 0x01FFFFFFu) | (2u << 30);  // addr[56:32] | type=2
  i32x8 g1;
  g1[0] = 0x00020000;            // data_size = 2 (4 bytes)
  g1[1] = (int)(4096u << 16);    // tensor_dim0[15:0]=0x1000 at bits[31:16]
  g1[2] = 0x00010000;            // tensor_dim0 hi=0 ; tensor_dim1=1
  g1[3] = (int)(4096u << 16);    // tile_dim0=0x1000 at bits[31:16]
  g1[4] = 1;                     // tile_dim1=1, tile_dim2=0
  g1[5] = 4096;                  // tensor_dim0_stride[31:0]
  g1[6] = 0;
  g1[7] = 0;
  i32x4 z4 = {};
#if __clang_major__ >= 23
  i32x8 z8 = {};
  __builtin_amdgcn_tensor_load_to_lds(g0, g1, z4, z4, z8, 0);
#else
  __builtin_amdgcn_tensor_load_to_lds(g0, g1, z4, z4, 0);
#endif
}

// ---------------------------------------------------------------------------
// ||M||_F reduction -> invn = eps / (||M|| + 1e-8)
// ---------------------------------------------------------------------------
__global__ __launch_bounds__(256) void tf_norm(const float* __restrict__ Mat,
                                               float* __restrict__ invn) {
  __shared__ float red[256];
  float s = 0.f;
  for (int i = threadIdx.x; i < Kk * Kk; i += 256) {
    float v = Mat[i];
    s += v * v;
  }
  red[threadIdx.x] = s;
  __syncthreads();
  for (int off = 128; off > 0; off >>= 1) {
    if ((int)threadIdx.x < off) red[threadIdx.x] += red[threadIdx.x + off];
    __syncthreads();
  }
  if (threadIdx.x == 0) invn[0] = 0.01f / (sqrtf(red[0]) + 1e-8f);
}

// R = I + invn * M   (unpadded 256x256 in workspace)
__global__ __launch_bounds__(256) void tf_buildR(const float* __restrict__ Mat,
                                                 const float* __restrict__ invn,
                                                 float* __restrict__ Rg) {
  int i = blockIdx.x, j = threadIdx.x;
  float v = invn[0] * Mat[i * Kk + j];
  if (i == j) v += 1.0f;
  Rg[i * Kk + j] = v;
}

// ---------------------------------------------------------------------------
// Phase 1: beta[t,b,k] = sum_d x[b,t,d] * W_in[k,d]
// One block per t (16 batch rows = one WMMA M tile), 8 waves x 2 N-tiles.
// ---------------------------------------------------------------------------
__global__ __launch_bounds__(256) void tf_proj_in(const float* __restrict__ x,
                                                  const float* __restrict__ Win,
                                                  float* __restrict__ beta) {
  const int s = blockIdx.x;  // time index
  const int tid = threadIdx.x;
  const int lane = tid & 31, wave = tid >> 5;
  const int ll = lane >> 4, lr = lane & 15;
  const int n0 = wave * 32, n1 = wave * 32 + 16;

  __shared__ float Ald[16 * APAD];
  __shared__ float Bld[32 * BPAD];

  v8f acc0 = {}, acc1 = {};

  for (int dbase = 0; dbase < Dd; dbase += 32) {
    // A tile: x[b][s][dbase..+32)  (b = row)
    {
      int r = tid >> 4, off = (tid & 15) * 2;
      const float* src = x + (size_t)r * Tt * Dd + (size_t)s * Dd + dbase + off;
      Ald[r * APAD + off] = src[0];
      Ald[r * APAD + off + 1] = src[1];
    }
    // B tile: B[d][k] = W_in[k][d]; coalesced float4 along d, transpose into LDS
    for (int rep = 0; rep < 8; ++rep) {
      int idx4 = rep * 256 + tid;      // 2048 float4 elements
      int kcol = idx4 >> 3;            // 0..255
      int dl4 = (idx4 & 7) * 4;        // 0,4,..28
      const float4 g = *(const float4*)(Win + (size_t)kcol * Dd + dbase + dl4);
      Bld[(dl4 + 0) * BPAD + kcol] = g.x;
      Bld[(dl4 + 1) * BPAD + kcol] = g.y;
      Bld[(dl4 + 2) * BPAD + kcol] = g.z;
      Bld[(dl4 + 3) * BPAD + kcol] = g.w;
    }
    // Hint the next chunk toward the caches while we compute this one.
    if (dbase + 32 < Dd) {
      __builtin_prefetch(Win + (size_t)tid * Dd + dbase + 32, 0, 1);
      __builtin_prefetch(x + (size_t)(tid & 15) * Tt * Dd + (size_t)s * Dd + dbase + 32, 0, 1);
    }
    __syncthreads();
#pragma unroll
    for (int kk = 0; kk < 32; kk += 4) {
      int kb = kk + 2 * ll;  // lane-half K offset
      v2f a = *(const v2f*)&Ald[lr * APAD + kb];
      v2f b0, b1;
      b0.x = Bld[(kb + 0) * BPAD + n0 + lr];
      b0.y = Bld[(kb + 1) * BPAD + n0 + lr];
      b1.x = Bld[(kb + 0) * BPAD + n1 + lr];
      b1.y = Bld[(kb + 1) * BPAD + n1 + lr];
      acc0 = wmma4(a, b0, acc0);
      acc1 = wmma4(a, b1, acc1);
    }
    __syncthreads();
  }
  // C layout: element (m,n) -> vgpr m&7, lane ((m>>3)<<4)|n
  float* outp = beta + (size_t)s * Bb * Kk;
#pragma unroll
  for (int v = 0; v < 8; ++v) {
    int m = v + 8 * ll;
    outp[m * Kk + n0 + lr] = acc0[v];
    outp[m * Kk + n1 + lr] = acc1[v];
  }
}

// ---------------------------------------------------------------------------
// Phase 2: serial scan. One workgroup, 16 waves; R resident in LDS; state
// (sr,si) lives in WMMA accumulator registers for all 2048 steps; beta_t is
// TDM-prefetched into LDS one step ahead, overlapping the WMMA phase.
// ---------------------------------------------------------------------------
__global__ __launch_bounds__(512) void tf_scan(const float* __restrict__ beta,
                                               const float* __restrict__ Rg,
                                               const float* __restrict__ alpha,
                                               const float* __restrict__ omega,
                                               float* __restrict__ srall,
                                               float* __restrict__ state_r,
                                               float* __restrict__ state_i) {
  extern __shared__ float lds[];
  float* Rl  = lds;                   // 256 * RPAD
  float* nrb = Rl + 256 * RPAD;       // 16 * NPAD
  float* nib = nrb + 16 * NPAD;       // 16 * NPAD
  float* cA  = nib + 16 * NPAD;       // 256 (mag*cos)
  float* cB  = cA + 256;              // 256 (mag*sin)
  float* btl = cB + 256;              // 4096 (beta_t staging, TDM target)

  const int tid = threadIdx.x;
  const int lane = tid & 31, wave = tid >> 5;
  const int ll = lane >> 4, lr = lane & 15;
  const int n = wave * 16 + lr;  // output column owned by this lane in C layout

  // Kick off the DMA of beta_0 before doing the R restride.
  if (wave == 0) tdm_load_row4096(beta, btl);

  // Prologue: restride R into LDS; fold rotation into 2 coefficient vectors.
  for (int i = tid; i < Kk * Kk; i += 512)
    Rl[(i >> 8) * RPAD + (i & 255)] = Rg[i];
  if (tid < Kk) {
    float mag = 0.99f / (1.f + __expf(-alpha[tid]));
    float w = omega[tid];
    cA[tid] = mag * __cosf(w);
    cB[tid] = mag * __sinf(w);
  }
  if (wave == 0) __builtin_amdgcn_s_wait_tensorcnt((unsigned short)0);
  __syncthreads();

  v8f cr = {}, ci = {};
  const float a = cA[n], b = cB[n];

  for (int t = 0; t < Tt; ++t) {
    // Elementwise rotation in C-register layout -> nr/ni exchange buffers.
    // beta_t comes from the TDM-staged LDS buffer.
#pragma unroll
    for (int v = 0; v < 8; ++v) {
      int m = v + 8 * ll;
      float sr = cr[v], si = ci[v];
      float bet = btl[m * Kk + n];
      nrb[m * NPAD + n] = a * sr - b * si + bet;
      nib[m * NPAD + n] = b * sr + a * si;
    }
    __syncthreads();

    // Everyone is done reading btl: start DMA of beta_{t+1}; it overlaps the
    // 128-WMMA matmul phase below and is drained before the closing barrier.
    if (wave == 0 && t + 1 < Tt)
      tdm_load_row4096(beta + (size_t)(t + 1) * Bb * Kk, btl);

    // sr' = nr @ R ; si' = ni @ R  (B fragment of R shared by both)
    v8f accr = {}, acci = {};
#pragma unroll 4
    for (int kk = 0; kk < Kk; kk += 4) {
      int kb = kk + 2 * ll;
      v2f ar = *(const v2f*)&nrb[lr * NPAD + kb];
      v2f ai = *(const v2f*)&nib[lr * NPAD + kb];
      v2f bf;
      bf.x = Rl[(kb + 0) * RPAD + n];
      bf.y = Rl[(kb + 1) * RPAD + n];
      accr = wmma4(ar, bf, accr);
      acci = wmma4(ai, bf, acci);
    }
    cr = accr;
    ci = acci;

    float* so = srall + (size_t)t * Bb * Kk;
#pragma unroll
    for (int v = 0; v < 8; ++v) so[(v + 8 * ll) * Kk + n] = accr[v];

    if (wave == 0) __builtin_amdgcn_s_wait_tensorcnt((unsigned short)0);
    __syncthreads();  // btl fully written + nrb/nib free for next step
  }

  // Final states (B=16, K=256)
#pragma unroll
  for (int v = 0; v < 8; ++v) {
    int m = v + 8 * ll;
    state_r[m * Kk + n] = cr[v];
    state_i[m * Kk + n] = ci[v];
  }
}

// ---------------------------------------------------------------------------
// Phase 3: out[b,t,d] = sum_k sr_all[t,b,k] * W_out[d,k]
// grid = (T, 4): one block per (t, 256-wide d slab)
// ---------------------------------------------------------------------------
__global__ __launch_bounds__(256) void tf_proj_out(const float* __restrict__ srall,
                                                   const float* __restrict__ Wout,
                                                   float* __restrict__ out) {
  const int s = blockIdx.x;      // time index
  const int dslab = blockIdx.y;  // 0..3
  const int tid = threadIdx.x;
  const int lane = tid & 31, wave = tid >> 5;
  const int ll = lane >> 4, lr = lane & 15;
  const int n0 = wave * 32, n1 = wave * 32 + 16;

  __shared__ float Ald[16 * APAD];
  __shared__ float Bld[32 * BPAD];

  v8f acc0 = {}, acc1 = {};

  for (int kbase = 0; kbase < Kk; kbase += 32) {
    // A tile: sr_all[s][b][kbase..+32)
    {
      int r = tid >> 4, off = (tid & 15) * 2;
      const float* src = srall + (size_t)s * Bb * Kk + r * Kk + kbase + off;
      Ald[r * APAD + off] = src[0];
      Ald[r * APAD + off + 1] = src[1];
    }
    // B tile: B[k][d] = W_out[d][k]; coalesced float4 along k, transpose to LDS
    for (int rep = 0; rep < 8; ++rep) {
      int idx4 = rep * 256 + tid;
      int dl = idx4 >> 3;              // 0..255 within slab
      int kl4 = (idx4 & 7) * 4;        // 0,4,..28
      const float4 g =
          *(const float4*)(Wout + (size_t)(dslab * 256 + dl) * Kk + kbase + kl4);
      Bld[(kl4 + 0) * BPAD + dl] = g.x;
      Bld[(kl4 + 1) * BPAD + dl] = g.y;
      Bld[(kl4 + 2) * BPAD + dl] = g.z;
      Bld[(kl4 + 3) * BPAD + dl] = g.w;
    }
    if (kbase + 32 < Kk)
      __builtin_prefetch(Wout + (size_t)(dslab * 256 + tid) * Kk + kbase + 32, 0, 1);
    __syncthreads();
#pragma unroll
    for (int kk = 0; kk < 32; kk += 4) {
      int kb = kk + 2 * ll;
      v2f aa = *(const v2f*)&Ald[lr * APAD + kb];
      v2f b0, b1;
      b0.x = Bld[(kb + 0) * BPAD + n0 + lr];
      b0.y = Bld[(kb + 1) * BPAD + n0 + lr];
      b1.x = Bld[(kb + 0) * BPAD + n1 + lr];
      b1.y = Bld[(kb + 1) * BPAD + n1 + lr];
      acc0 = wmma4(aa, b0, acc0);
      acc1 = wmma4(aa, b1, acc1);
    }
    __syncthreads();
  }
  // out[b][s][dslab*256 + n]
#pragma unroll
  for (int v = 0; v < 8; ++v) {
    int m = v + 8 * ll;
    float* op = out + (size_t)m * Tt * Dd + (size_t)s * Dd + dslab * 256;
    op[n0 + lr] = acc0[v];
    op[n1 + lr] = acc1[v];
  }
}

// ---------------------------------------------------------------------------
extern "C" void kernel_launch(void* const* d_in, const int* in_sizes, int n_in,
                              void* d_out, int out_size, void* d_ws,
                              size_t ws_size, hipStream_t stream) {
  (void)in_sizes; (void)n_in; (void)out_size; (void)ws_size;
  const float* x     = (const float*)d_in[0];
  const float* alpha = (const float*)d_in[1];
  const float* omega = (const float*)d_in[2];
  const float* Win   = (const float*)d_in[3];
  const float* Wout  = (const float*)d_in[4];
  const float* Mat   = (const float*)d_in[5];

  float* out = (float*)d_out;
  float* ws = (float*)d_ws;

  float* beta  = ws;                                  // T*B*K floats
  float* srall = beta + (size_t)Tt * Bb * Kk;         // T*B*K floats
  float* Rg    = srall + (size_t)Tt * Bb * Kk;        // K*K floats
  float* invn  = Rg + (size_t)Kk * Kk;                // 1 float

  float* outMain = out;
  float* stR = out + (size_t)Bb * Tt * Dd;
  float* stI = stR + (size_t)Bb * Kk;

  tf_norm<<<1, 256, 0, stream>>>(Mat, invn);
  tf_buildR<<<Kk, Kk, 0, stream>>>(Mat, invn, Rg);
  tf_proj_in<<<Tt, 256, 0, stream>>>(x, Win, beta);

  // LDS: R (256*264) + nr/ni (2*16*260) + coeffs (512) + beta stage (4096)
  size_t smem =
      (size_t)(256 * RPAD + 2 * 16 * NPAD + 2 * 256 + Bb * Kk) * sizeof(float);
  tf_scan<<<1, 512, smem, stream>>>(beta, Rg, alpha, omega, srall, stR, stI);

  tf_proj_out<<<dim3(Tt, 4), 256, 0, stream>>>(srall, Wout, outMain);
}